// GraphSAGE_85255100826267
// MI455X (gfx1250) — compile-verified
//
#include <hip/hip_runtime.h>
#include <hip/hip_bf16.h>

typedef __attribute__((ext_vector_type(2))) float v2f;
typedef __attribute__((ext_vector_type(8))) float v8f;

#define IN_FEATS 128
#define HID      128
#define NCLS     64
#define N3 819200
#define N2 204800
#define N1 40960
#define N0 4096
#define E0 (N2 * 10)
#define E1 (N1 * 10)
#define E2 (N0 * 10)

// ---------------------------------------------------------------------------
// Zero-fill (grid-stride)
// ---------------------------------------------------------------------------
__global__ void zero_kernel(float* __restrict__ p, long long n) {
    long long i = (long long)blockIdx.x * blockDim.x + threadIdx.x;
    long long stride = (long long)gridDim.x * blockDim.x;
    for (; i < n; i += stride) p[i] = 0.0f;
}

// ---------------------------------------------------------------------------
// Edge-parallel mean-aggregation (sum part): one wave32 per edge.
// Each lane loads a float4 (global_load_b128; 512B per wave, fully coalesced)
// from the src row and atomic-adds into the dst row. Lane 0 bumps degree.
// ---------------------------------------------------------------------------
__global__ void edge_aggregate(const float* __restrict__ feat,
                               const int*   __restrict__ src,
                               const int*   __restrict__ dst,
                               float* __restrict__ agg,
                               float* __restrict__ deg,
                               int nedges) {
    int e = blockIdx.x * (blockDim.x >> 5) + (threadIdx.x >> 5);
    if (e >= nedges) return;                    // wave-uniform exit
    int lane = threadIdx.x & 31;
    int s = src[e];
    int d = dst[e];
    const float4* f4 = (const float4*)(feat + (size_t)s * 128);
    float*        a  = agg + (size_t)d * 128;
    float4 v = f4[lane];
    int c = lane * 4;
    atomicAdd(&a[c + 0], v.x);                  // global_atomic_add_f32
    atomicAdd(&a[c + 1], v.y);
    atomicAdd(&a[c + 2], v.z);
    atomicAdd(&a[c + 3], v.w);
    if (lane == 0) atomicAdd(&deg[d], 1.0f);
}

// ---------------------------------------------------------------------------
// h_neigh = agg / max(deg, 1)   (feature width fixed at 128)
// ---------------------------------------------------------------------------
__global__ void normalize_agg(float* __restrict__ agg,
                              const float* __restrict__ deg,
                              long long n) {
    long long i = (long long)blockIdx.x * blockDim.x + threadIdx.x;
    if (i >= n) return;
    float dd = deg[i >> 7];
    agg[i] = agg[i] / fmaxf(dd, 1.0f);
}

// ---------------------------------------------------------------------------
// Fused SAGE dense stage using V_WMMA_F32_16X16X4_F32:
//   Out[m][:] = Hself[m]·Wself + Hneigh[m]·Wneigh + bias   (optional ReLU)
// One wave owns a 16-row strip across ALL N columns (NTILES 16x16 tiles kept
// in registers): each A fragment is loaded once and feeds NTILES WMMAs, so A
// traffic drops NTILES-fold vs tile-per-wave. W stays L2-resident (<=64KB).
// A 16x4 f32 layout: lanes 0-15 row M=l hold K={k0,k0+1}; lanes 16-31 hold
// K={k0+2,k0+3}. B 4x16 mirrored. C/D: VGPR r -> row (r + 8*half), lane->col.
// ---------------------------------------------------------------------------
template <int NTILES, int RELU>
__global__ void sage_gemm_wmma(const float* __restrict__ Hself,   // M x 128
                               const float* __restrict__ Hneigh,  // M x 128
                               const float* __restrict__ Wself,   // 128 x N
                               const float* __restrict__ Wneigh,  // 128 x N
                               const float* __restrict__ bias,    // N
                               float* __restrict__ Out,           // M x N
                               int M) {
    const int K = 128;
    const int N = NTILES * 16;
    int strip = blockIdx.x * (blockDim.x >> 5) + (threadIdx.x >> 5);
    if (strip >= (M >> 4)) return;              // wave-uniform exit
    int m0 = strip << 4;
    int lane = threadIdx.x & 31;
    int half = lane >> 4;                       // 0: K pair {0,1}; 1: {2,3}
    int l    = lane & 15;

    v8f acc[NTILES];
#pragma unroll
    for (int t = 0; t < NTILES; ++t) acc[t] = (v8f){};

    const float* arow_s = Hself  + (size_t)(m0 + l) * K;
    const float* arow_n = Hneigh + (size_t)(m0 + l) * K;

    // Pass 1: Hself x Wself
#pragma unroll 2
    for (int k0 = 0; k0 < K; k0 += 4) {
        int ka = k0 + 2 * half;
        v2f a;
        a.x = arow_s[ka];
        a.y = arow_s[ka + 1];
        const float* b0 = Wself + ka * N + l;        // 32-bit offsets only
        const float* b1 = Wself + (ka + 1) * N + l;
#pragma unroll
        for (int t = 0; t < NTILES; ++t) {
            v2f b;
            b.x = b0[t * 16];
            b.y = b1[t * 16];
            acc[t] = __builtin_amdgcn_wmma_f32_16x16x4_f32(
                         /*neg_a=*/false, a, /*neg_b=*/false, b,
                         /*c_mod=*/(short)0, acc[t],
                         /*reuse_a=*/false, /*reuse_b=*/false);
        }
    }
    // Pass 2: Hneigh x Wneigh
#pragma unroll 2
    for (int k0 = 0; k0 < K; k0 += 4) {
        int ka = k0 + 2 * half;
        v2f a;
        a.x = arow_n[ka];
        a.y = arow_n[ka + 1];
        const float* b0 = Wneigh + ka * N + l;
        const float* b1 = Wneigh + (ka + 1) * N + l;
#pragma unroll
        for (int t = 0; t < NTILES; ++t) {
            v2f b;
            b.x = b0[t * 16];
            b.y = b1[t * 16];
            acc[t] = __builtin_amdgcn_wmma_f32_16x16x4_f32(
                         false, a, false, b, (short)0, acc[t], false, false);
        }
    }

    // Epilogue: bias (+ReLU), write 16 x N strip.
#pragma unroll
    for (int t = 0; t < NTILES; ++t) {
        float bv = bias[t * 16 + l];
#pragma unroll
        for (int r = 0; r < 8; ++r) {
            float v = acc[t][r] + bv;
            if (RELU) v = fmaxf(v, 0.0f);
            Out[(size_t)(m0 + r + 8 * half) * N + t * 16 + l] = v;
        }
    }
}

// ---------------------------------------------------------------------------
// Host-side pipeline
// ---------------------------------------------------------------------------
static inline void launch_zero(float* p, long long n, hipStream_t s) {
    int blocks = (int)((n + 255) / 256);
    if (blocks > 65535) blocks = 65535;
    zero_kernel<<<blocks, 256, 0, s>>>(p, n);
}

static inline void launch_agg(const float* feat, const int* src, const int* dst,
                              float* agg, float* deg, int E, hipStream_t s) {
    int waves_per_block = 8;                    // 256 threads
    int blocks = (E + waves_per_block - 1) / waves_per_block;
    edge_aggregate<<<blocks, 256, 0, s>>>(feat, src, dst, agg, deg, E);
}

static inline void launch_norm(float* agg, const float* deg, long long n, hipStream_t s) {
    normalize_agg<<<(int)((n + 255) / 256), 256, 0, s>>>(agg, deg, n);
}

template <int NTILES, int RELU>
static inline void launch_gemm(const float* Hs, const float* Hn,
                               const float* Ws, const float* Wn, const float* b,
                               float* out, int M, hipStream_t s) {
    int strips = M >> 4;                        // one wave per 16-row strip
    int blocks = (strips + 7) / 8;              // 8 waves per 256-thread block
    sage_gemm_wmma<NTILES, RELU><<<blocks, 256, 0, s>>>(Hs, Hn, Ws, Wn, b, out, M);
}

extern "C" void kernel_launch(void* const* d_in, const int* in_sizes, int n_in,
                              void* d_out, int out_size, void* d_ws, size_t ws_size,
                              hipStream_t stream) {
    const float* x_l1 = (const float*)d_in[0];
    const float* x_l0 = (const float*)d_in[1];
    const int* src0 = (const int*)d_in[2];
    const int* dst0 = (const int*)d_in[3];
    const int* src1 = (const int*)d_in[4];
    const int* dst1 = (const int*)d_in[5];
    const int* src2 = (const int*)d_in[6];
    const int* dst2 = (const int*)d_in[7];
    const float* Ws0 = (const float*)d_in[8];
    const float* Wn0 = (const float*)d_in[9];
    const float* b0  = (const float*)d_in[10];
    const float* Ws1 = (const float*)d_in[11];
    const float* Wn1 = (const float*)d_in[12];
    const float* b1  = (const float*)d_in[13];

    float* out = (float*)d_out;
    float* h_seed = out;                     // N0 x 64
    float* h_neib = out + (size_t)N0 * NCLS; // N1 x 64

    // Workspace carve (floats)
    float* ws = (float*)d_ws;
    float* agg0 = ws;                         // N2*128
    float* deg0 = agg0 + (size_t)N2 * 128;    // N2   (contiguous with agg0)
    float* h2   = deg0 + N2;                  // N2*128
    float* agg1 = h2 + (size_t)N2 * 128;      // N1*128
    float* deg1 = agg1 + (size_t)N1 * 128;    // N1
    float* h1   = deg1 + N1;                  // N1*128
    float* agg2 = h1 + (size_t)N1 * 128;      // N0*128
    float* deg2 = agg2 + (size_t)N0 * 128;    // N0

    // --- Layer-0 conv on edges0: x_l1 -> h2 (N2 x 128, ReLU) ---
    launch_zero(agg0, (long long)N2 * 128 + N2, stream);
    launch_agg(x_l1, src0, dst0, agg0, deg0, E0, stream);
    launch_norm(agg0, deg0, (long long)N2 * 128, stream);
    launch_gemm<8, 1>(x_l1, agg0, Ws0, Wn0, b0, h2, N2, stream);

    // --- h_neib = sage_conv(h2, edges1, W1) (N1 x 64, no ReLU) ---
    launch_zero(agg1, (long long)N1 * 128 + N1, stream);
    launch_agg(h2, src1, dst1, agg1, deg1, E1, stream);
    launch_norm(agg1, deg1, (long long)N1 * 128, stream);
    launch_gemm<4, 0>(h2, agg1, Ws1, Wn1, b1, h_neib, N1, stream);

    // --- Layer-0 conv on edges1: x_l0 -> h1 (N1 x 128, ReLU) ---
    launch_zero(agg1, (long long)N1 * 128 + N1, stream);
    launch_agg(x_l0, src1, dst1, agg1, deg1, E1, stream);
    launch_norm(agg1, deg1, (long long)N1 * 128, stream);
    launch_gemm<8, 1>(x_l0, agg1, Ws0, Wn0, b0, h1, N1, stream);

    // --- h_seed = sage_conv(h1, edges2, W1) (N0 x 64, no ReLU) ---
    launch_zero(agg2, (long long)N0 * 128 + N0, stream);
    launch_agg(h1, src2, dst2, agg2, deg2, E2, stream);
    launch_norm(agg2, deg2, (long long)N0 * 128, stream);
    launch_gemm<4, 0>(h1, agg2, Ws1, Wn1, b1, h_seed, N0, stream);
}